// TemporalShift_71442486002134
// MI455X (gfx1250) — compile-verified
//
#include <hip/hip_runtime.h>
#include <stdint.h>

// TemporalShift: out[n, 0:16]   = mask_next[n] ? x[n+1, 0:16]  : 0
//                out[n, 16:32]  = mask_prev[n] ? x[n-1, 16:32] : 0
//                out[n, 32:128] = x[n, 32:128]
//                out[n] = x[n] wherever skip[n]
// Pure permutation, 822 MB total traffic -> HBM-bound (~35 us @ 23.3 TB/s).
// Strategy: b128 NT streaming loads/stores, block-uniform mask scalarization.

typedef __attribute__((ext_vector_type(4))) float v4f;

#define NN    1024
#define CC    128
#define HW4   196            // float4s per (n,c) plane: 28*28/4
#define PER_N (CC * HW4)     // 25088 float4s per sample
#define FOLD  16             // C / SHIFT_DIV

__global__ __launch_bounds__(256)
void TemporalShift_71442486002134_kernel(const v4f* __restrict__ x,
                                         const unsigned char* __restrict__ mask_next,
                                         const unsigned char* __restrict__ mask_prev,
                                         const unsigned char* __restrict__ skip,
                                         v4f* __restrict__ out)
{
    const int n = blockIdx.y;                              // sample index, block-uniform
    const int i = blockIdx.x * 256 + threadIdx.x;          // 0..PER_N-1 (exact, no tail)
    const int c = i / HW4;                                 // channel (magic-mul division)

    // Block-uniform predicate loads -> compiler scalarizes to s_load.
    const bool sk = skip[n] != 0;

    int  src  = n;      // source sample for the load
    bool zero = false;  // write zeros instead of loading
    if (!sk) {
        if (c < FOLD) {                    // shift-left group: take from n+1
            if (mask_next[n]) src = n + 1; else zero = true;
        } else if (c < 2 * FOLD) {         // shift-right group: take from n-1
            if (mask_prev[n]) src = n - 1; else zero = true;
        }
        // c >= 32: pass-through, src = n
    }

    v4f v;
    if (zero) {
        v = (v4f)(0.0f);
    } else {
        // Streamed exactly once; dataset (411 MB) >> L2 (192 MB) -> non-temporal.
        v = __builtin_nontemporal_load(&x[(long long)src * PER_N + i]);
    }
    __builtin_nontemporal_store(v, &out[(long long)n * PER_N + i]);
}

extern "C" void kernel_launch(void* const* d_in, const int* in_sizes, int n_in,
                              void* d_out, int out_size, void* d_ws, size_t ws_size,
                              hipStream_t stream)
{
    // setup_inputs() order: x (f32), mask_next (bool/u8), mask_prev (bool/u8), skip (bool/u8)
    const v4f*           x  = (const v4f*)d_in[0];
    const unsigned char* mn = (const unsigned char*)d_in[1];
    const unsigned char* mp = (const unsigned char*)d_in[2];
    const unsigned char* sk = (const unsigned char*)d_in[3];
    v4f*                 o  = (v4f*)d_out;

    dim3 grid(PER_N / 256, NN, 1);   // 98 x 1024 blocks, 256 threads (8 wave32)
    TemporalShift_71442486002134_kernel<<<grid, dim3(256, 1, 1), 0, stream>>>(x, mn, mp, sk, o);
}